// PesqLoss_23356032155740
// MI455X (gfx1250) — compile-verified
//
#include <hip/hip_runtime.h>
#include <math.h>
#include <stdint.h>

#ifndef M_PI
#define M_PI 3.14159265358979323846
#endif

// ---------------- problem constants ----------------
#define B_SZ   32
#define T_LEN  163840
#define NFFT   512
#define HOP    256
#define NB     49
#define NBIN   257
#define NF     639          // (T - NFFT)/HOP + 1
#define MT     40           // ceil(NF/16)
#define NPAD   272          // 257 freq bins padded to 17 WMMA N-tiles
#define CHUNK  (15*HOP + NFFT)   // 4352 contiguous samples per 16-frame tile
#define SP_C   6.910853e-6f
#define SL_C   1.866055e-1f
#define ZW_C   0.23f

typedef float v2f __attribute__((ext_vector_type(2)));
typedef float v8f __attribute__((ext_vector_type(8)));
typedef unsigned int v4u __attribute__((ext_vector_type(4)));
typedef int v8i __attribute__((ext_vector_type(8)));
typedef int v4i __attribute__((ext_vector_type(4)));

// ---------------- workspace layout (float offsets) ----------------
static constexpr size_t OFF_BUTTER = 0;                                   // b[11], a[11]
static constexpr size_t OFF_THRESH = 32;                                  // 49
static constexpr size_t OFF_THPOW  = 96;                                  // 49: (TH/0.5)^ZW
static constexpr size_t OFF_BSTART = 704;                                 // 50 ints
static constexpr size_t OFF_MX     = 768;                                 // 32
static constexpr size_t OFF_WIDTH  = 800;                                 // 1
static constexpr size_t OFF_COS    = 1024;                                // 256 K-pairs * 272 * 2 (window folded in)
static constexpr size_t OFF_SIN    = OFF_COS + (size_t)NFFT * NPAD;
static constexpr size_t OFF_Y      = OFF_SIN + (size_t)NFFT * NPAD;       // 2*B*T
static constexpr size_t OFF_BANDS  = OFF_Y + (size_t)2 * B_SZ * T_LEN;    // 2*B*NF*NB
static constexpr size_t OFF_BPR    = OFF_BANDS + (size_t)2 * B_SZ * NF * NB;
static constexpr size_t OFF_SYMM   = OFF_BPR + (size_t)B_SZ * NB + 16;
static constexpr size_t OFF_ASYM   = OFF_SYMM + (size_t)B_SZ * NF;

// ---------------- complex double helpers (Butterworth design) ----------------
struct cd { double x, y; };
__device__ __forceinline__ cd cadd(cd a, cd b) { return {a.x + b.x, a.y + b.y}; }
__device__ __forceinline__ cd csub(cd a, cd b) { return {a.x - b.x, a.y - b.y}; }
__device__ __forceinline__ cd cmul(cd a, cd b) { return {a.x*b.x - a.y*b.y, a.x*b.y + a.y*b.x}; }
__device__ __forceinline__ cd cscale(cd a, double s) { return {a.x*s, a.y*s}; }
__device__ __forceinline__ cd cdivd(cd a, cd b) {
    double d = b.x*b.x + b.y*b.y;
    return {(a.x*b.x + a.y*b.y)/d, (a.y*b.x - a.x*b.y)/d};
}
__device__ __forceinline__ cd csqrtd(cd z) {
    double r = sqrt(z.x*z.x + z.y*z.y);
    double u = sqrt(0.5 * (r + z.x));
    double v = sqrt(fmax(0.5 * (r - z.x), 0.0));
    if (z.y < 0.0) v = -v;
    return {u, v};
}
__device__ void polyroots(const cd* r, int n, cd* c) {
    c[0] = {1.0, 0.0};
    for (int j = 1; j <= n; ++j) c[j] = {0.0, 0.0};
    for (int i = 0; i < n; ++i)
        for (int j = n; j >= 1; --j)
            c[j] = csub(c[j], cmul(r[i], c[j-1]));
}

// ---------------- K0: scalar constant tables ----------------
__global__ void k_init_scalar(float* ws) {
    if (threadIdx.x != 0) return;

    // Bark band map + thresholds
    double zmax = 7.0 * asinh(8000.0 / 650.0);
    double step = zmax / (double)NB;
    ws[OFF_WIDTH] = (float)step;
    int bidx[NBIN];
    for (int i = 0; i < NBIN; ++i) {
        double f = (double)i * 16000.0 / (double)NFFT;
        double z = 7.0 * asinh(f / 650.0);
        int b = (int)floor(z / step);
        if (b < 0) b = 0;
        if (b > NB - 1) b = NB - 1;
        bidx[i] = b;
    }
    int* bstart = (int*)(ws + OFF_BSTART);
    {
        int cur = 0;
        for (int b = 0; b < NB; ++b) {
            while (cur < NBIN && bidx[cur] < b) cur++;
            bstart[b] = cur;
        }
        bstart[NB] = NBIN;
    }
    for (int b = 0; b < NB; ++b) {
        double zc = step * ((double)b + 0.5);
        double fk = fmax(650.0 * sinh(zc / 7.0), 50.0) / 1000.0;
        double thr_db = 3.64 * pow(fk, -0.8)
                      - 6.5 * exp(-0.6 * (fk - 3.3) * (fk - 3.3))
                      + 1e-3 * pow(fk, 4.0);
        double th = pow(10.0, thr_db / 10.0);
        ws[OFF_THRESH + b] = (float)th;
        ws[OFF_THPOW  + b] = (float)pow(th / 0.5, (double)ZW_C);
    }

    // Butterworth bandpass order 5, 300..3500 Hz @ 16 kHz (scipy butter equivalent)
    const int ORD = 5;
    cd p[ORD];
    for (int k = 1; k <= ORD; ++k) {
        double th = M_PI * (2.0*k + ORD - 1.0) / (2.0*ORD);
        p[k-1] = {cos(th), sin(th)};
    }
    double fs2 = 32000.0;
    double wlo = fs2 * tan(M_PI * 300.0  / 16000.0);
    double whi = fs2 * tan(M_PI * 3500.0 / 16000.0);
    double bw  = whi - wlo, wo = sqrt(wlo * whi);
    cd pbp[10];
    for (int i = 0; i < ORD; ++i) {
        cd plp = cscale(p[i], bw * 0.5);
        cd rad = csqrtd(csub(cmul(plp, plp), cd{wo*wo, 0.0}));
        pbp[i]       = cadd(plp, rad);
        pbp[i + ORD] = csub(plp, rad);
    }
    double gain = pow(bw, (double)ORD);
    cd pd[10];
    for (int i = 0; i < 10; ++i)
        pd[i] = cdivd(cadd(cd{fs2, 0.0}, pbp[i]), csub(cd{fs2, 0.0}, pbp[i]));
    cd den = {1.0, 0.0};
    for (int i = 0; i < 10; ++i) den = cmul(den, csub(cd{fs2, 0.0}, pbp[i]));
    cd g = cdivd(cd{pow(fs2, (double)ORD), 0.0}, den);
    gain *= g.x;
    cd zd[10], bc[11], ac[11];
    for (int i = 0; i < ORD; ++i) { zd[i] = {1.0, 0.0}; zd[i + ORD] = {-1.0, 0.0}; }
    polyroots(zd, 10, bc);
    polyroots(pd, 10, ac);
    for (int j = 0; j < 11; ++j) {
        ws[OFF_BUTTER + j]      = (float)(bc[j].x * gain);
        ws[OFF_BUTTER + 11 + j] = (float)(ac[j].x);
    }
}

// ---------------- K1: window-folded DFT basis, packed as K-pairs for b64 loads ------
// layout: basis[((k>>1)*NPAD + n)*2 + (k&1)] = hann(k) * cos/sin(2*pi*k*n/512)
__global__ void k_init_basis(float* ws) {
    long total = (long)NFFT * NPAD;
    for (long i = (long)blockIdx.x * blockDim.x + threadIdx.x; i < total;
         i += (long)gridDim.x * blockDim.x) {
        int k = (int)(i / NPAD), n = (int)(i % NPAD);
        double w = 0.5 * (1.0 - cos(2.0 * M_PI * (double)k / (double)NFFT));
        double ang = 2.0 * M_PI * (double)k * (double)n / (double)NFFT;
        size_t dst = ((size_t)(k >> 1) * NPAD + n) * 2 + (k & 1);
        ws[OFF_COS + dst] = (float)(w * cos(ang));
        ws[OFF_SIN + dst] = (float)(w * sin(ang));
    }
}

// ---------------- K2: per-row max(|ref|,|deg|) ----------------
__global__ void k_maxabs(const float* __restrict__ ref, const float* __restrict__ deg,
                         float* ws) {
    __shared__ float red[256];
    int b = blockIdx.x, tid = threadIdx.x;
    const float* r = ref + (size_t)b * T_LEN;
    const float* d = deg + (size_t)b * T_LEN;
    float m = 0.f;
    for (int t = tid; t < T_LEN; t += 256) {
        m = fmaxf(m, fabsf(r[t]));
        m = fmaxf(m, fabsf(d[t]));
    }
    red[tid] = m; __syncthreads();
    for (int s = 128; s > 0; s >>= 1) {
        if (tid < s) red[tid] = fmaxf(red[tid], red[tid + s]);
        __syncthreads();
    }
    if (tid == 0) ws[OFF_MX + b] = red[0];
}

// ---------------- K3: level align (order-10 IIR power) + pre-emphasis biquad --------
// Strictly sequential recurrences -> one lane per (signal,row); 64-way parallel.
__global__ void k_filter(const float* __restrict__ ref, const float* __restrict__ deg,
                         float* ws) {
    int row = blockIdx.x;                 // 0..2B-1
    if (threadIdx.x != 0) return;
    int sig = row / B_SZ, b = row % B_SZ;
    const float* x = (sig == 0 ? ref : deg) + (size_t)b * T_LEN;
    float* y = ws + OFF_Y + (size_t)row * T_LEN;
    float inv_mx = 1.0f / ws[OFF_MX + b];

    float bb[11], aa[11];
#pragma unroll
    for (int i = 0; i < 11; ++i) { bb[i] = ws[OFF_BUTTER + i]; aa[i] = ws[OFF_BUTTER + 11 + i]; }

    // pass 1: DF2T bandpass, accumulate filtered power
    float z[10];
#pragma unroll
    for (int i = 0; i < 10; ++i) z[i] = 0.f;
    double acc = 0.0;
    for (int t = 0; t < T_LEN; ++t) {
        if ((t & 1023) == 0) __builtin_prefetch(x + t + 4096, 0, 1);
        float xt = x[t] * inv_mx;
        float yt = bb[0] * xt + z[0];
#pragma unroll
        for (int i = 0; i < 9; ++i) z[i] = bb[i+1]*xt - aa[i+1]*yt + z[i+1];
        z[9] = bb[10]*xt - aa[10]*yt;
        acc += (double)yt * (double)yt;
    }
    double power = acc / (double)(T_LEN + 5120) / 1.04684;
    float scale = (float)sqrt(1e7 / power) * inv_mx;

    // pass 2: edge taper + pre-emphasis biquad (DF2T)
    const float pb0 = 2.740826f, pb1 = -5.4816519f, pb2 = 2.740826f;
    const float pa1 = -1.9444777f, pa2 = 0.94597794f;
    float z0 = 0.f, z1 = 0.f;
    for (int t = 0; t < T_LEN; ++t) {
        if ((t & 1023) == 0) __builtin_prefetch(x + t + 4096, 0, 1);
        float xt = x[t] * scale;
        if (t < 15)          xt *= (float)(t + 1) * (1.0f / 16.0f);
        if (t >= T_LEN - 15) xt *= (float)(T_LEN - t) * (1.0f / 16.0f);
        float yt = pb0 * xt + z0;
        z0 = pb1 * xt - pa1 * yt + z1;
        z1 = pb2 * xt - pa2 * yt;
        y[t] = yt;
    }
}

// ---------------- K4: spectrogram: TDM tile stage + fp32 WMMA DFT-GEMM --------------
// One wave per 16-frame tile. A = raw contiguous sample chunk DMA'd by the Tensor
// Data Mover with 4-dword padding every 256 dwords (LDS row stride 260 ==> banks
// 4*lane mod 64: conflict-free fragment reads). Hann window is folded into the basis.
__global__ void __launch_bounds__(32) k_spec(float* ws) {
    __shared__ float Ach[4448];          // 4352 data + 17*4 pad dwords
    __shared__ float Pt[16][NPAD + 4];
    int blk = blockIdx.x;
    int row = blk / MT;                  // sig*B + b
    int mtile = blk % MT;
    int m0 = mtile * 16;
    const float* y  = ws + OFF_Y + (size_t)row * T_LEN;
    const float* CB = ws + OFF_COS;
    const float* SB = ws + OFF_SIN;
    int tid = threadIdx.x;

    // ---- Tensor Data Mover: 1D tile, data_size=4B, pad 4 dwords per 256 dwords.
    {
        unsigned lds_off = (unsigned)(uintptr_t)(void*)&Ach[0];
        unsigned long long ga = (unsigned long long)(uintptr_t)(y + (size_t)m0 * HOP);
        unsigned remain = (unsigned)(T_LEN - m0 * HOP);   // OOB tail reads return 0
        v4u g0;
        g0[0] = 1u;                                               // count=1, user mode
        g0[1] = lds_off;                                          // lds_addr
        g0[2] = (unsigned)ga;                                     // global_addr[95:64]
        g0[3] = (unsigned)((ga >> 32) & 0x01FFFFFFull) | (2u << 30); // addr[120:96], type=2
        v8i g1;
        g1[0] = (int)((2u << 16) | (1u << 20) | (7u << 22) | (3u << 25));
                 // data_size=4B, pad_enable, pad_interval=256dw, pad_amount=4dw
        g1[1] = (int)((remain & 0xFFFFu) << 16);                  // tensor_dim0 lo16
        g1[2] = (int)((remain >> 16) | (1u << 16));               // tensor_dim0 hi16, tensor_dim1=1
        g1[3] = (int)(((unsigned)CHUNK) << 16);                   // tile_dim0=4352
        g1[4] = 1;                                                // tile_dim1=1, tile_dim2=0
        g1[5] = (int)T_LEN;                                       // tensor_dim0_stride (unused, 1D)
        g1[6] = 0;
        g1[7] = 0;
        v4i gz4; gz4[0] = 0; gz4[1] = 0; gz4[2] = 0; gz4[3] = 0;
        v8i gz8; gz8[0] = 0; gz8[1] = 0; gz8[2] = 0; gz8[3] = 0;
                 gz8[4] = 0; gz8[5] = 0; gz8[6] = 0; gz8[7] = 0;
        __builtin_amdgcn_tensor_load_to_lds(g0, g1, gz4, gz4, gz8, 0);
        __builtin_amdgcn_s_wait_tensorcnt(0);
    }
    __syncthreads();

    int lh = tid >> 4;                   // half-wave select (K offset 0 / +2)
    int ll = tid & 15;
    int koff = lh << 1;
    const int arow = ll * 260;           // padded row stride: 256 data + 4 pad dwords

    for (int nt = 0; nt < 17; ++nt) {
        int n0 = nt * 16;
        v8f cc = {};
        v8f cs = {};
        const float* cb = CB + 2 * (n0 + ll);
        const float* sb = SB + 2 * (n0 + ll);
        for (int kk = 0; kk < 128; ++kk) {
            int k = (kk << 2) + koff;
            int kadj = k + ((k & 256) ? 4 : 0);          // skip mid-chunk pad
            v2f a = *(const v2f*)&Ach[arow + kadj];
            int kp = (kk << 1) + lh;
            v2f bc = *(const v2f*)&cb[(size_t)kp * (NPAD * 2)];
            v2f bs = *(const v2f*)&sb[(size_t)kp * (NPAD * 2)];
            cc = __builtin_amdgcn_wmma_f32_16x16x4_f32(false, a, false, bc, (short)0, cc, false, false);
            cs = __builtin_amdgcn_wmma_f32_16x16x4_f32(false, a, false, bs, (short)0, cs, false, false);
        }
        int mbase = lh * 8;
        int n = n0 + ll;
#pragma unroll
        for (int r = 0; r < 8; ++r) {
            float pw = cc[r] * cc[r] + cs[r] * cs[r];
            Pt[mbase + r][n] = (n == 0) ? 0.f : pw;      // zero DC bin
        }
    }
    __syncthreads();

    // deterministic segmented Bark-band reduction (bands are contiguous bin ranges)
    const int* bstart = (const int*)(ws + OFF_BSTART);
    float* bands = ws + OFF_BANDS;
    for (int i = tid; i < 16 * NB; i += 32) {
        int mf = i / NB, band = i % NB;
        int fr = m0 + mf;
        if (fr >= NF) continue;
        float s = 0.f;
        int e = bstart[band + 1];
        for (int bin = bstart[band]; bin < e; ++bin) s += Pt[mf][bin];
        bands[((size_t)row * NF + fr) * NB + band] = SP_C * s;
    }
}

// ---------------- K5: silent-frame mask, time-avg audible, band power ratio ---------
__global__ void k_avg_bpr(float* ws) {
    __shared__ unsigned char sil[NF];
    int b = blockIdx.x, tid = threadIdx.x;
    const float* TH   = ws + OFF_THRESH;
    const float* refb = ws + OFF_BANDS + (size_t)b * NF * NB;
    const float* degb = ws + OFF_BANDS + (size_t)(B_SZ + b) * NF * NB;

    for (int f = tid; f < NF; f += blockDim.x) {
        const float* r = refb + (size_t)f * NB;
        float s = 0.f;
        for (int k = 0; k < NB; ++k) {
            float v = r[k];
            if (v > TH[k] * 100.f) s += v;
        }
        sil[f] = (s < 1e7f) ? 1 : 0;
    }
    __syncthreads();
    if (tid >= NB) return;

    float thk = TH[tid] * 100.f;
    float ar = 0.f, ad = 0.f; int ns = 0;
    for (int f = 0; f < NF; ++f) {
        if (!sil[f]) {
            ns++;
            float rv = refb[(size_t)f * NB + tid]; if (rv > thk) ar += rv;
            float dv = degb[(size_t)f * NB + tid]; if (dv > thk) ad += dv;
        }
    }
    float denom = (float)(ns > 0 ? ns : 1);
    float bpr = (ad / denom + 1000.f) / (ar / denom + 1000.f);
    ws[OFF_BPR + (size_t)b * NB + tid] = fminf(fmaxf(bpr, 0.01f), 100.f);
}

// ---------------- K6: per-frame disturbance (loudness, symm/asym norms) -------------
__global__ void k_frame(float* ws) {
    int idx = blockIdx.x * blockDim.x + threadIdx.x;
    if (idx >= B_SZ * NF) return;
    int b = idx / NF, f = idx % NF;
    const float* TH   = ws + OFF_THRESH;
    const float* THP  = ws + OFF_THPOW;
    const float* bpr  = ws + OFF_BPR + (size_t)b * NB;
    const float* refb = ws + OFF_BANDS + ((size_t)b * NF + f) * NB;
    const float* degb = ws + OFF_BANDS + ((size_t)(B_SZ + b) * NF + f) * NB;

    float ta_er = 0.f, ta_d = 0.f;
    for (int k = 0; k < NB; ++k) {
        float er = bpr[k] * refb[k]; if (er > TH[k]) ta_er += er;
        float dv = degb[k];          if (dv > TH[k]) ta_d  += dv;
    }
    float fc = (ta_er + 5000.f) / (ta_d + 5000.f);
    float fpr = fc;
    if (f > 0) {
        const float* rp = refb - NB;
        const float* dp = degb - NB;
        float per = 0.f, pdd = 0.f;
        for (int k = 0; k < NB; ++k) {
            float er = bpr[k] * rp[k]; if (er > TH[k]) per += er;
            float dv = dp[k];          if (dv > TH[k]) pdd += dv;
        }
        float fp = (per + 5000.f) / (pdd + 5000.f);
        fpr = 0.8f * fc + 0.2f * fp;
    }
    fpr = fminf(fmaxf(fpr, 0.0003f), 5.f);

    float s2 = 0.f, s1 = 0.f;
    for (int k = 0; k < NB; ++k) {
        float th = TH[k];
        float er = bpr[k] * refb[k];
        float ed = fpr * degb[k];
        float dl = (ed > th) ? SL_C * THP[k] * (powf(0.5f + 0.5f * ed / th, ZW_C) - 1.f) : 0.f;
        float rl = (er > th) ? SL_C * THP[k] * (powf(0.5f + 0.5f * er / th, ZW_C) - 1.f) : 0.f;
        float dead = 0.25f * fminf(dl, rl);
        float dist = dl - rl;
        float mag  = fmaxf(fabsf(dist) - dead, 0.f);
        dist = copysignf(mag, dist);
        s2 += dist * dist;
        float asc = powf((ed + 50.f) / (er + 50.f), 1.2f);
        asc = (asc < 3.f) ? 0.f : fminf(asc, 12.f);
        s1 += fabsf(dist) * asc;
    }
    float Wd = ws[OFF_WIDTH];
    float symm = fmaxf(sqrtf(Wd * s2), 1e-20f);
    float asym = fmaxf(Wd * s1, 1e-20f);
    float h = powf((ta_er + 1e5f) * 1e-7f, 0.04f);
    symm = fminf(symm / h, 45.f);
    asym = fminf(asym / h, 45.f);
    ws[OFF_SYMM + (size_t)b * NF + f] = symm;
    ws[OFF_ASYM + (size_t)b * NF + f] = asym;
}

// ---------------- K7: PSQM windows + final score --------------------------------
__global__ void k_psqm(const float* __restrict__ ws, float* __restrict__ out) {
    __shared__ float rs[64], ra[64];
    int b = blockIdx.x, tid = threadIdx.x;
    const float* sm = ws + OFF_SYMM + (size_t)b * NF;
    const float* am = ws + OFF_ASYM + (size_t)b * NF;
    const int NW = (NF - 20) / 10 + 1;   // 62 windows
    float ss = 0.f, sa = 0.f;
    for (int w = tid; w < NW; w += 64) {
        float m6s = 0.f, m6a = 0.f;
        for (int j = 0; j < 20; ++j) {
            float vs = sm[w * 10 + j], va = am[w * 10 + j];
            float v2 = vs * vs; m6s += v2 * v2 * v2;
            float a2 = va * va; m6a += a2 * a2 * a2;
        }
        m6s *= (1.f / 20.f);  m6a *= (1.f / 20.f);
        float ps = powf(m6s, 1.f / 6.f);
        float pa = powf(m6a, 1.f / 6.f);
        ss += ps * ps;  sa += pa * pa;
    }
    rs[tid] = ss; ra[tid] = sa; __syncthreads();
    if (tid == 0) {
        float ts = 0.f, ta = 0.f;
        for (int i = 0; i < 64; ++i) { ts += rs[i]; ta += ra[i]; }
        float ds = sqrtf(ts / (float)NW);
        float da = sqrtf(ta / (float)NW);
        out[b] = 0.5f * (0.1f * ds + 0.0309f * da);
    }
}

// ---------------- launcher ----------------
extern "C" void kernel_launch(void* const* d_in, const int* in_sizes, int n_in,
                              void* d_out, int out_size, void* d_ws, size_t ws_size,
                              hipStream_t stream) {
    (void)in_sizes; (void)n_in; (void)out_size; (void)ws_size;
    const float* ref = (const float*)d_in[0];
    const float* deg = (const float*)d_in[1];
    float* ws  = (float*)d_ws;
    float* out = (float*)d_out;

    k_init_scalar<<<1, 64, 0, stream>>>(ws);
    k_init_basis <<<272, 512, 0, stream>>>(ws);
    k_maxabs     <<<B_SZ, 256, 0, stream>>>(ref, deg, ws);
    k_filter     <<<2 * B_SZ, 32, 0, stream>>>(ref, deg, ws);
    k_spec       <<<2 * B_SZ * MT, 32, 0, stream>>>(ws);
    k_avg_bpr    <<<B_SZ, 64, 0, stream>>>(ws);
    k_frame      <<<(B_SZ * NF + 255) / 256, 256, 0, stream>>>(ws);
    k_psqm       <<<B_SZ, 64, 0, stream>>>(ws, out);
}